// GlobalSpatialCorrelationLoss_10763188043930
// MI455X (gfx1250) — compile-verified
//
#include <hip/hip_runtime.h>
#include <hip/hip_bf16.h>
#include <math.h>

// Problem constants (from reference)
#define BATCH 16
#define CHW   65536         // C*H*W
#define NSAMP 8192          // N samples
#define NTIL  512           // NSAMP / 16
#define NT    (NTIL * (NTIL + 1) / 2)   // 131328 lower-tri (incl diag) 16x16 tiles
#define EPSF  1e-12f

typedef __attribute__((ext_vector_type(2))) float v2f;
typedef __attribute__((ext_vector_type(8))) float v8f;

__global__ void gscl_init_acc(double* __restrict__ acc) {
    if (threadIdx.x < 8) acc[threadIdx.x] = 0.0;
}

// Gather + row-normalize features; gather coords and squared norms.
__global__ void gscl_prep(const float* __restrict__ feat,
                          const float* __restrict__ coords,
                          const int* __restrict__ sidx,
                          float* __restrict__ Xn,
                          float* __restrict__ cg,
                          float* __restrict__ qn) {
    int n = blockIdx.x * blockDim.x + threadIdx.x;
    if (n >= NSAMP) return;
    int id = sidx[n];
    float x[BATCH];
    float mean = 0.f;
#pragma unroll
    for (int b = 0; b < BATCH; ++b) {
        x[b] = feat[(size_t)b * CHW + id];
        mean += x[b];
    }
    mean *= (1.f / BATCH);
    float ss = 0.f;
#pragma unroll
    for (int b = 0; b < BATCH; ++b) {
        x[b] -= mean;
        ss += x[b] * x[b];
    }
    float inv = 1.f / sqrtf(ss);   // Xn = Xc / sqrt(sum(Xc^2))
#pragma unroll
    for (int b = 0; b < BATCH; ++b)
        Xn[(size_t)n * BATCH + b] = x[b] * inv;
    float cx = coords[(size_t)id * 2 + 0];
    float cy = coords[(size_t)id * 2 + 1];
    cg[2 * n + 0] = cx;
    cg[2 * n + 1] = cy;
    qn[n] = cx * cx + cy * cy;
}

// One wave per 16x16 tile (it >= jt). WMMA f32 16x16x4 Gram + distance sim,
// branchless masked accumulation into double accumulators.
__global__ void gscl_tiles(const float* __restrict__ Xn,
                           const float* __restrict__ cg,
                           const float* __restrict__ qn,
                           double* __restrict__ acc) {
    int wave = threadIdx.x >> 5;
    int lane = threadIdx.x & 31;
    int t = blockIdx.x * (blockDim.x >> 5) + wave;
    if (t >= NT) return;                     // uniform per wave: EXEC stays all-1s

    // decode triangular index: t = it*(it+1)/2 + jt, jt <= it
    int it = (int)((sqrtf(8.0f * (float)t + 1.0f) - 1.0f) * 0.5f);
    while ((it + 1) * (it + 2) / 2 <= t) ++it;
    while (it * (it + 1) / 2 > t) --it;
    int jt = t - it * (it + 1) / 2;
    int i0 = it << 4, j0 = jt << 4;

    int lane16 = lane & 15;
    int hihalf = lane >> 4;                  // 0 or 1

    // A 16x4 layout: lanes 0-15 M=0..15 {K=k,k+1}; lanes 16-31 {K=k+2,k+3}.
    // B 4x16 layout mirrors it for a Gram matrix: same per-lane gather from row tile j0.
    const float* __restrict__ Xa = Xn + (size_t)(i0 + lane16) * BATCH;
    const float* __restrict__ Xb = Xn + (size_t)(j0 + lane16) * BATCH;
    int kb = hihalf * 2;

    v8f c = {};
#pragma unroll
    for (int kk = 0; kk < 4; ++kk) {
        v2f a = *(const v2f*)(Xa + 4 * kk + kb);
        v2f b = *(const v2f*)(Xb + 4 * kk + kb);
        // D = A x B + C  (fp32, exact-precision Gram accumulation)
        c = __builtin_amdgcn_wmma_f32_16x16x4_f32(
                /*neg_a=*/false, a, /*neg_b=*/false, b,
                /*c_mod=*/(short)0, c, /*reuse_a=*/false, /*reuse_b=*/false);
    }

    // Per-lane own-row coordinate data; i-rows broadcast via wave shuffles.
    int j = j0 + lane16;
    float xj = cg[2 * j + 0], yj = cg[2 * j + 1], qj = qn[j];
    int irow = i0 + lane16;                  // row lane16 of the i-tile
    float xr = cg[2 * irow + 0], yr = cg[2 * irow + 1], qr = qn[irow];
    bool offdiag = (it != jt);

    float sa = 0.f, sb = 0.f, saa = 0.f, sbb = 0.f, sab = 0.f;
#pragma unroll
    for (int v = 0; v < 8; ++v) {
        int m8 = hihalf * 8 + v;             // row within tile (0..15)
        int i = i0 + m8;
        // broadcast row m8's coords from lane m8 (rows live on lanes 0..15)
        float xi = __shfl(xr, m8, 32);
        float yi = __shfl(yr, m8, 32);
        float qi = __shfl(qr, m8, 32);

        float aR = c[v];
        float dot = fmaf(yi, yj, xi * xj);
        float sq = fmaf(-2.f, dot, qi + qj);
        // max(sq,0) then where(sq>EPS, sq, EPS) == max(sq, EPS)
        float d  = __builtin_amdgcn_sqrtf(fmaxf(sq, EPSF));
        float bD = __builtin_amdgcn_rcpf(d + 1.f);

        float m  = (offdiag || (i > j)) ? 1.f : 0.f;   // strict lower triangle
        float aRm = aR * m;
        float bDm = bD * m;
        sa  += aRm;        sb  += bDm;
        saa += aRm * aR;   sbb += bDm * bD;   sab += aRm * bD;
    }

    // wave32 butterfly reduction
#pragma unroll
    for (int off = 16; off > 0; off >>= 1) {
        sa  += __shfl_xor(sa,  off, 32);
        sb  += __shfl_xor(sb,  off, 32);
        saa += __shfl_xor(saa, off, 32);
        sbb += __shfl_xor(sbb, off, 32);
        sab += __shfl_xor(sab, off, 32);
    }
    if (lane == 0) {
        unsafeAtomicAdd(&acc[0], (double)sa);
        unsafeAtomicAdd(&acc[1], (double)sb);
        unsafeAtomicAdd(&acc[2], (double)saa);
        unsafeAtomicAdd(&acc[3], (double)sbb);
        unsafeAtomicAdd(&acc[4], (double)sab);
    }
}

__global__ void gscl_finalize(const double* __restrict__ acc, float* __restrict__ out) {
    double cnt = (double)NSAMP * (double)(NSAMP - 1) * 0.5;
    double sa = acc[0], sb = acc[1], saa = acc[2], sbb = acc[3], sab = acc[4];
    double am = sa / cnt, bm = sb / cnt;
    double cov = sab - cnt * am * bm;
    double va  = saa - cnt * am * am;
    double vb  = sbb - cnt * bm * bm;
    double r   = cov / (sqrt(va) * sqrt(vb));
    out[0] = (float)((1.0 - r) * 0.5);
}

extern "C" void kernel_launch(void* const* d_in, const int* in_sizes, int n_in,
                              void* d_out, int out_size, void* d_ws, size_t ws_size,
                              hipStream_t stream) {
    const float* feat   = (const float*)d_in[0];   // [16, 64, 32, 32] fp32
    const float* coords = (const float*)d_in[1];   // [65536, 2] fp32
    const int*   sidx   = (const int*)d_in[2];     // [8192] int
    float* out = (float*)d_out;

    char* ws = (char*)d_ws;
    double* acc = (double*)ws;                                   // 64 B
    float* Xn = (float*)(ws + 256);                              // 8192*16 f32 = 512 KB
    float* cg = Xn + (size_t)NSAMP * BATCH;                      // 8192*2  f32
    float* qn = cg + (size_t)NSAMP * 2;                          // 8192    f32

    gscl_init_acc<<<1, 32, 0, stream>>>(acc);
    gscl_prep<<<NSAMP / 256, 256, 0, stream>>>(feat, coords, sidx, Xn, cg, qn);

    const int wavesPerBlock = 8;                                 // 256 thr = 8 wave32
    int blocks = (NT + wavesPerBlock - 1) / wavesPerBlock;       // 16416
    gscl_tiles<<<blocks, 256, 0, stream>>>(Xn, cg, qn, acc);

    gscl_finalize<<<1, 1, 0, stream>>>(acc, out);
}